// SE_Module_72808285601940
// MI455X (gfx1250) — compile-verified
//
#include <hip/hip_runtime.h>
#include <math.h>

typedef __attribute__((ext_vector_type(2))) float v2f;
typedef __attribute__((ext_vector_type(8))) float v8f;

#define Bsz 32
#define Hsz 112
#define Wsz 112
#define Csz 128
#define Rsz 2048
#define HW (Hsz * Wsz)          // 12544
#define NCHUNK 32
#define CHPIX (HW / NCHUNK)     // 392 pixels per chunk

// ws layout (float offsets)
#define WS_PARTIAL 0
#define WS_S  (Bsz * NCHUNK * Csz)   // 131072
#define WS_H  (WS_S + Bsz * Csz)     // +4096
#define WS_G  (WS_H + Bsz * Rsz)     // +65536
// total floats = 204800 -> ~800 KB of ws

// ---------------------------------------------------------------------------
// Stage 1: partial spatial sums. grid=(NCHUNK, B), block=256 (8 waves).
// thread t: channel group c4 = (t&31)*4, spatial row-in-chunk j = t>>5.
// Each iteration a wave loads 32 float4 = 512 contiguous bytes (NHWC layout).
// ---------------------------------------------------------------------------
__global__ void se_pool_partial(const float* __restrict__ x, float* __restrict__ ws) {
    const int t  = threadIdx.x;
    const int b  = blockIdx.y;
    const int k  = blockIdx.x;
    const int cg = t & 31;        // channel float4 group 0..31
    const int j  = t >> 5;        // 0..7
    const float4* x4 = (const float4*)x;
    const long base = (long)(b * HW + k * CHPIX) * (Csz / 4) + cg;

    float4 acc = make_float4(0.f, 0.f, 0.f, 0.f);
    for (int p = j; p < CHPIX; p += 8) {
        float4 v = x4[base + (long)p * (Csz / 4)];
        acc.x += v.x; acc.y += v.y; acc.z += v.z; acc.w += v.w;
    }

    __shared__ float4 red[8][32];
    red[j][cg] = acc;
    __syncthreads();
    if (j == 0) {
        float4 s = red[0][cg];
        #pragma unroll
        for (int r = 1; r < 8; ++r) {
            float4 v = red[r][cg];
            s.x += v.x; s.y += v.y; s.z += v.z; s.w += v.w;
        }
        float4* outp = (float4*)(ws + WS_PARTIAL);
        outp[(b * NCHUNK + k) * (Csz / 4) + cg] = s;
    }
}

// ---------------------------------------------------------------------------
// Stage 2: fold chunks -> s[B][C], scale by 1/(H*W). 16 blocks x 256.
// ---------------------------------------------------------------------------
__global__ void se_pool_reduce(float* __restrict__ ws) {
    const int t = blockIdx.x * blockDim.x + threadIdx.x;
    if (t < Bsz * Csz) {
        const int b = t >> 7;
        const int c = t & 127;
        const float* p = ws + WS_PARTIAL + b * NCHUNK * Csz + c;
        float s = 0.f;
        #pragma unroll 8
        for (int k = 0; k < NCHUNK; ++k) s += p[k * Csz];
        ws[WS_S + t] = s * (1.0f / (float)HW);
    }
}

__device__ __forceinline__ float gelu_tanh(float v) {
    const float k0 = 0.7978845608028654f;  // sqrt(2/pi)
    const float k1 = 0.044715f;
    float inner = k0 * (v + k1 * v * v * v);
    return 0.5f * v * (1.0f + tanhf(inner));
}

// ---------------------------------------------------------------------------
// Stage 3: h = gelu(s @ w1 + b1), s[32,128] w1[128,2048] -> h[32,2048].
// One 16x16 output tile per wave via V_WMMA_F32_16X16X4_F32, K stepped by 4.
// grid=32 blocks x 8 waves = 256 waves = 2 row-tiles x 128 col-tiles.
// Fragment layouts per ISA 7.12.2 (fp32, wave32):
//   A 16x4: lane<16 -> M=lane, K = {0,1}; lane>=16 -> M=lane-16, K = {2,3}
//   B 4x16: lane<16 -> N=lane, K = {0,1}; lane>=16 -> N=lane-16, K = {2,3}
//   C/D   : VGPR v, lane l -> M = v + (l>=16 ? 8 : 0), N = l&15
// ---------------------------------------------------------------------------
__global__ void se_mm1(const float* __restrict__ w1, const float* __restrict__ b1,
                       float* __restrict__ ws) {
    const int lane = threadIdx.x & 31;
    const int wave = threadIdx.x >> 5;
    const int W    = blockIdx.x * 8 + wave;  // 0..255
    const int mt   = W >> 7;                 // row tile 0..1
    const int nt   = W & 127;                // col tile 0..127
    const int rowc = lane & 15;              // M for A-frag / N for B-frag
    const int kl   = (lane >> 4) * 2;        // K sub-offset {0,2}

    const float* s = ws + WS_S;
    v8f acc = {};
    for (int kk = 0; kk < Csz / 4; ++kk) {
        const int k0 = kk * 4 + kl;
        v2f a, bf;
        a.x  = s[(mt * 16 + rowc) * Csz + k0];
        a.y  = s[(mt * 16 + rowc) * Csz + k0 + 1];
        bf.x = w1[k0 * Rsz + nt * 16 + rowc];
        bf.y = w1[(k0 + 1) * Rsz + nt * 16 + rowc];
        acc = __builtin_amdgcn_wmma_f32_16x16x4_f32(
            /*neg_a=*/false, a, /*neg_b=*/false, bf,
            /*c_mod=*/(short)0, acc, /*reuse_a=*/false, /*reuse_b=*/false);
    }

    const int n     = nt * 16 + (lane & 15);
    const int mbase = mt * 16 + ((lane >> 4) * 8);
    const float bias = b1[n];
    float* h = ws + WS_H;
    #pragma unroll
    for (int v = 0; v < 8; ++v) {
        float val = acc[v] + bias;
        h[(mbase + v) * Rsz + n] = gelu_tanh(val);
    }
}

// ---------------------------------------------------------------------------
// Stage 4: g = sigmoid(h @ w2 + b2), h[32,2048] w2[2048,128] -> g[32,128].
// 2 blocks x 8 waves = 16 waves = 2 row-tiles x 8 col-tiles, K=2048 (512 steps).
// ---------------------------------------------------------------------------
__global__ void se_mm2(const float* __restrict__ w2, const float* __restrict__ b2,
                       float* __restrict__ ws) {
    const int lane = threadIdx.x & 31;
    const int wave = threadIdx.x >> 5;
    const int W    = blockIdx.x * 8 + wave;  // 0..15
    const int mt   = W >> 3;                 // 0..1
    const int nt   = W & 7;                  // 0..7
    const int rowc = lane & 15;
    const int kl   = (lane >> 4) * 2;

    const float* h = ws + WS_H;
    v8f acc = {};
    for (int kk = 0; kk < Rsz / 4; ++kk) {
        const int k0 = kk * 4 + kl;
        v2f a, bf;
        a.x  = h[(mt * 16 + rowc) * Rsz + k0];
        a.y  = h[(mt * 16 + rowc) * Rsz + k0 + 1];
        bf.x = w2[k0 * Csz + nt * 16 + rowc];
        bf.y = w2[(k0 + 1) * Csz + nt * 16 + rowc];
        acc = __builtin_amdgcn_wmma_f32_16x16x4_f32(
            false, a, false, bf, (short)0, acc, false, false);
    }

    const int n     = nt * 16 + (lane & 15);
    const int mbase = mt * 16 + ((lane >> 4) * 8);
    const float bias = b2[n];
    float* g = ws + WS_G;
    #pragma unroll
    for (int v = 0; v < 8; ++v) {
        float val = acc[v] + bias;
        g[(mbase + v) * Csz + n] = 1.0f / (1.0f + __expf(-val));
    }
}

// ---------------------------------------------------------------------------
// Stage 5: out = x * g (gate broadcast over H,W). Pure streaming float4.
// grid=(HW*C/4/256, B) = (1568, 32) blocks x 256 threads.
// Per pixel there are 32 float4 channel groups -> group index = i & 31.
// ---------------------------------------------------------------------------
__global__ void se_scale(const float* __restrict__ x, const float* __restrict__ ws,
                         float* __restrict__ out) {
    const int b = blockIdx.y;
    const int i = blockIdx.x * blockDim.x + threadIdx.x;   // float4 idx in image
    const long idx = (long)b * (HW * Csz / 4) + i;
    const float4* x4 = (const float4*)x;
    const float4* g4 = (const float4*)(ws + WS_G);
    float4* o4 = (float4*)out;

    float4 xv = x4[idx];
    float4 gv = g4[b * (Csz / 4) + (i & 31)];
    float4 r;
    r.x = xv.x * gv.x; r.y = xv.y * gv.y; r.z = xv.z * gv.z; r.w = xv.w * gv.w;
    o4[idx] = r;
}

extern "C" void kernel_launch(void* const* d_in, const int* in_sizes, int n_in,
                              void* d_out, int out_size, void* d_ws, size_t ws_size,
                              hipStream_t stream) {
    const float* x  = (const float*)d_in[0];
    const float* w1 = (const float*)d_in[1];
    const float* b1 = (const float*)d_in[2];
    const float* w2 = (const float*)d_in[3];
    const float* b2 = (const float*)d_in[4];
    float* out = (float*)d_out;
    float* ws  = (float*)d_ws;

    se_pool_partial<<<dim3(NCHUNK, Bsz), 256, 0, stream>>>(x, ws);
    se_pool_reduce<<<16, 256, 0, stream>>>(ws);
    se_mm1<<<32, 256, 0, stream>>>(w1, b1, ws);
    se_mm2<<<2, 256, 0, stream>>>(w2, b2, ws);
    se_scale<<<dim3((HW * Csz / 4) / 256, Bsz), 256, 0, stream>>>(x, ws, out);
}